// Bdfdv_51170240364850
// MI455X (gfx1250) — compile-verified
//
#include <hip/hip_runtime.h>
#include <hip/hip_bf16.h>

#define L1C 5
#define NXC 1024
#define NVC 512

// ---------------------------------------------------------------------------
// CDNA5 async HBM->LDS path (ASYNCcnt). Guarded: falls back to regular
// NT loads + LDS stores if the builtin is missing on this toolchain.
// ---------------------------------------------------------------------------
#if __has_builtin(__builtin_amdgcn_global_load_async_to_lds_b128)
#define HAS_ASYNC_LDS 1
#else
#define HAS_ASYNC_LDS 0
#endif

typedef float v4f __attribute__((ext_vector_type(4)));

__device__ __forceinline__ v4f nt_load4(const float* p) {
    return __builtin_nontemporal_load(reinterpret_cast<const v4f*>(p));
}
__device__ __forceinline__ void nt_store4(float* p, v4f v) {
    __builtin_nontemporal_store(v, reinterpret_cast<v4f*>(p));
}
// Store 4 complex values (re/im interleaved) as two b128 NT stores.
__device__ __forceinline__ void store_cplx4(float* p, v4f re, v4f im) {
    v4f lo = {re.x, im.x, re.y, im.y};
    v4f hi = {re.z, im.z, re.w, im.w};
    nt_store4(p, lo);
    nt_store4(p + 4, hi);
}

#if HAS_ASYNC_LDS
// Builtin signature (from hipcc diagnostic): param0 is
// 'int __attribute__((vector_size(16))) __device__ *'  i.e. int4 in AS(1);
// param1 is the LDS-side pointer (AS(3)).
typedef int v4i __attribute__((vector_size(16)));
typedef __attribute__((address_space(1))) v4i* gv4i_p;  // pointee in global AS
typedef __attribute__((address_space(3))) v4i* lv4i_p;  // pointee in LDS AS

__device__ __forceinline__ void async_ld_b128(const float* g, v4f* lds) {
    // per-lane b128 HBM -> LDS, tracked by ASYNCcnt, no VGPR data movement
    __builtin_amdgcn_global_load_async_to_lds_b128(
        (gv4i_p)(v4i*)const_cast<float*>(g),
        (lv4i_p)(v4i*)lds,
        /*offset=*/0, /*cpol=*/0);
}
__device__ __forceinline__ void wait_async0() {
#if __has_builtin(__builtin_amdgcn_s_wait_asynccnt)
    __builtin_amdgcn_s_wait_asynccnt(0);
#else
    asm volatile("s_wait_asynccnt 0x0" ::: "memory");
#endif
}
#endif

// One block per (l, x): 128 threads x 4 v-values = 512 v.
// F rows 0..l are staged once into LDS (async ASYNCcnt-tracked copies when
// available), then all three stencil terms read them from LDS.
__global__ __launch_bounds__(128)
void bdfdv_kernel(const float* __restrict__ f_re,
                  const float* __restrict__ f_im,
                  const float* __restrict__ d0_re,
                  const float* __restrict__ d0_im,
                  const float* __restrict__ b,
                  float* __restrict__ out)
{
    __shared__ v4f sm_re[L1C][128];   // 10 KB
    __shared__ v4f sm_im[L1C][128];   // 10 KB

    const int bid = (int)blockIdx.x;
    const int l = bid >> 10;          // / NXC
    const int x = bid & (NXC - 1);    // % NXC
    const int t = (int)threadIdx.x;
    const int iv = t << 2;

    // b coefficients: uniform across the block -> scalar loads (s_load_b96)
    const float bx = b[3 * x + 0];
    const float b1 = b[3 * x + 1];   // bm = b1 + i*b2 ; bp = b1 - i*b2
    const float b2 = b[3 * x + 2];

    const size_t mstride = (size_t)NXC * NVC;
    // element offset of (l, m=0, x, iv); row m adds m*mstride
    const size_t base = (size_t)(l * L1C) * mstride + (size_t)x * NVC + (size_t)iv;

    // D0 is read-once and private to this thread: direct NT register load,
    // issued before the staging wait so it overlaps the async copies.
    const v4f D0r = nt_load4(d0_re + base);
    const v4f D0i = nt_load4(d0_im + base);

    // ---- stage F rows 0..l into LDS (only rows this l actually uses) ----
    if (l >= 1) {                      // uniform per block
        const int nr = l + 1;
        for (int r = 0; r < nr; ++r) {
            const size_t off = base + (size_t)r * mstride;
#if HAS_ASYNC_LDS
            async_ld_b128(f_re + off, &sm_re[r][t]);
            async_ld_b128(f_im + off, &sm_im[r][t]);
#else
            sm_re[r][t] = nt_load4(f_re + off);
            sm_im[r][t] = nt_load4(f_im + off);
#endif
        }
#if HAS_ASYNC_LDS
        wait_async0();
#endif
        __syncthreads();
    }

    const v4f vzero = {0.f, 0.f, 0.f, 0.f};
    const float lf = (float)l;

    // ---- m = 0 : D = D0 + cB*bm*F[l,1] + Re(cC*bp*F[l,1]) ----
    {
        v4f D_re = D0r;
        v4f D_im = D0i;
        if (l >= 1) {
            const v4f F1r = sm_re[1][t];
            const v4f F1i = sm_im[1][t];
            const float cB = -0.5f * lf * (lf + 1.0f);   // -(l-0)(l+0+1)/2
            D_re += cB * (b1 * F1r - b2 * F1i);
            D_im += cB * (b1 * F1i + b2 * F1r);
            const float cC = -lf * (lf + 1.0f);          // -l(l+1)
            // Re(cC * (b1 - i b2) * F1) = cC*(b1*F1re + b2*F1im)
            D_re += cC * (b1 * F1r + b2 * F1i);
        }
        store_cplx4(out + 2 * base, D_re, D_im);
    }

    // ---- m = 1..4 ----
    #pragma unroll
    for (int m = 1; m < L1C; ++m) {
        float* op = out + 2 * (base + (size_t)m * mstride);
        if (m > l) {
            // invalid (m > l): output must be exactly zero
            store_cplx4(op, vzero, vzero);
            continue;
        }
        const v4f Fpr = sm_re[m - 1][t], Fpi = sm_im[m - 1][t];
        const v4f Fcr = sm_re[m][t],     Fci = sm_im[m][t];
        // D = 0.5 * bm * F[l, m-1]
        v4f D_re = 0.5f * (b1 * Fpr - b2 * Fpi);
        v4f D_im = 0.5f * (b1 * Fpi + b2 * Fpr);
        // term A: -i * m * bx * F[l, m]
        const float a = (float)m * bx;
        D_re += a * Fci;
        D_im -= a * Fcr;
        if (m < l) {
            const v4f Fnr = sm_re[m + 1][t], Fni = sm_im[m + 1][t];
            const float mf = (float)m;
            const float cB = -0.5f * (lf - mf) * (lf + mf + 1.0f);
            D_re += cB * (b1 * Fnr - b2 * Fni);
            D_im += cB * (b1 * Fni + b2 * Fnr);
        }
        store_cplx4(op, D_re, D_im);
    }
}

extern "C" void kernel_launch(void* const* d_in, const int* in_sizes, int n_in,
                              void* d_out, int out_size, void* d_ws, size_t ws_size,
                              hipStream_t stream) {
    (void)in_sizes; (void)n_in; (void)out_size; (void)d_ws; (void)ws_size;
    const float* f_re  = (const float*)d_in[0];
    const float* f_im  = (const float*)d_in[1];
    const float* d0_re = (const float*)d_in[2];
    const float* d0_im = (const float*)d_in[3];
    const float* b     = (const float*)d_in[4];
    // d_in[5] = v : unused by the forward computation
    float* out = (float*)d_out;   // complex64 -> interleaved re/im float32

    dim3 grid(L1C * NXC);   // one block per (l, x)
    dim3 block(128);        // 4 wave32 waves; 128 * 4 v-lanes = NV
    bdfdv_kernel<<<grid, block, 0, stream>>>(f_re, f_im, d0_re, d0_im, b, out);
}